// Dihedral_19104014532721
// MI455X (gfx1250) — compile-verified
//
#include <hip/hip_runtime.h>

#define THREADS 256

// 12-byte position record; aligned(4) so the compiler can emit global_load_b96.
struct __attribute__((aligned(4))) F3 { float x, y, z; };

__device__ __forceinline__ F3 f3sub(F3 a, F3 b) { return {a.x - b.x, a.y - b.y, a.z - b.z}; }
__device__ __forceinline__ F3 f3cross(F3 a, F3 b) {
    return { a.y * b.z - a.z * b.y,
             a.z * b.x - a.x * b.z,
             a.x * b.y - a.y * b.x };
}
__device__ __forceinline__ float f3dot(F3 a, F3 b) {
    return fmaf(a.x, b.x, fmaf(a.y, b.y, a.z * b.z));
}

__global__ __launch_bounds__(THREADS) void dihedral_pot_kernel(
    const float* __restrict__ pos,       // [N,3]
    const float* __restrict__ th0_tab,   // [26^4]
    const float* __restrict__ k0_tab,    // [26^4]
    const float* __restrict__ th1_tab,   // [26^4]
    const float* __restrict__ k1_tab,    // [26^4]
    const int*   __restrict__ map,       // [4,M]
    const int*   __restrict__ types,     // [N]
    float* __restrict__ out,             // [M]
    int M)
{
    int m = blockIdx.x * THREADS + threadIdx.x;
    if (m >= M) return;

    // Speculative prefetch of the streamed mapping rows far ahead
    // (gfx1250 global_prefetch_b8). Branch-free: clamp instead of diverge —
    // prefetching our own (already-fetched) address at the tail is harmless.
    const int PF = 1 << 16;
    int pf = m + PF;
    pf = (pf < M) ? pf : m;
    __builtin_prefetch(map + 0 * M + pf, 0, 1);
    __builtin_prefetch(map + 1 * M + pf, 0, 1);
    __builtin_prefetch(map + 2 * M + pf, 0, 1);
    __builtin_prefetch(map + 3 * M + pf, 0, 1);

    // Coalesced, non-temporal streaming loads of the 4 atom indices.
    // NT keeps this one-shot 64MB stream from evicting the L2-resident
    // pos/type/param working set (~40MB << 192MB L2).
    int i0 = __builtin_nontemporal_load(map + 0 * M + m);
    int i1 = __builtin_nontemporal_load(map + 1 * M + m);
    int i2 = __builtin_nontemporal_load(map + 2 * M + m);
    int i3 = __builtin_nontemporal_load(map + 3 * M + m);

    // Issue all gathers up front so their L2 latencies overlap.
    int t0 = types[i0];
    int t1 = types[i1];
    int t2 = types[i2];
    int t3 = types[i3];

    const F3* __restrict__ p = reinterpret_cast<const F3*>(pos);
    F3 p0 = p[i0];
    F3 p1 = p[i1];
    F3 p2 = p[i2];
    F3 p3 = p[i3];

    // Parameter gathers (tables total 7.3MB — hot in L2 after first touch).
    int flat = ((t0 * 26 + t1) * 26 + t2) * 26 + t3;
    float th0 = th0_tab[flat];
    float kk0 = k0_tab[flat];
    float th1 = th1_tab[flat];
    float kk1 = k1_tab[flat];

    // Dihedral geometry.
    F3 b1 = f3sub(p1, p0);
    F3 b2 = f3sub(p2, p1);
    F3 b3 = f3sub(p3, p2);
    F3 n1 = f3cross(b1, b2);
    F3 n2 = f3cross(b2, b3);

    float x = f3dot(n1, n2);                                   // n1 . n2
    float y = f3dot(f3cross(n1, b2), n2)
              * __builtin_amdgcn_rsqf(f3dot(b2, b2));          // (n1 x b2n) . n2

    // theta = atan2(y, x); we only ever need cos(theta - th0) and
    // cos(2*theta - th1), so expand trigonometrically — no atan2 needed:
    //   cos t = x/r, sin t = y/r,  r = sqrt(x^2 + y^2)
    float rinv = __builtin_amdgcn_rsqf(fmaf(x, x, y * y));     // v_rsq_f32
    float ct = x * rinv;
    float st = y * rinv;
    float c2t = fmaf(ct, ct, -(st * st));                      // cos 2t
    float s2t = 2.0f * ct * st;                                // sin 2t

    // th0, th1 are uniform in [0,1): hardware v_sin/v_cos are near-exact there.
    float c0 = __cosf(th0), s0 = __sinf(th0);
    float c1 = __cosf(th1), s1 = __sinf(th1);

    // cos(t - th0) = ct*c0 + st*s0 ; cos(2t - th1) = c2t*c1 + s2t*s1
    float V = kk0 * (1.0f - fmaf(ct, c0, st * s0))
            + kk1 * (1.0f - fmaf(c2t, c1, s2t * s1));

    __builtin_nontemporal_store(V, out + m);
}

extern "C" void kernel_launch(void* const* d_in, const int* in_sizes, int n_in,
                              void* d_out, int out_size, void* d_ws, size_t ws_size,
                              hipStream_t stream) {
    const float* pos     = (const float*)d_in[0];
    const float* th0_tab = (const float*)d_in[1];
    const float* k0_tab  = (const float*)d_in[2];
    const float* th1_tab = (const float*)d_in[3];
    const float* k1_tab  = (const float*)d_in[4];
    const int*   mapping = (const int*)d_in[5];
    const int*   types   = (const int*)d_in[6];
    float*       out     = (float*)d_out;

    int M = in_sizes[5] / 4;   // mapping is [4, M]
    int blocks = (M + THREADS - 1) / THREADS;
    dihedral_pot_kernel<<<blocks, THREADS, 0, stream>>>(
        pos, th0_tab, k0_tab, th1_tab, k1_tab, mapping, types, out, M);
}